// TopicDecoder_50903952392722
// MI455X (gfx1250) — compile-verified
//
#include <hip/hip_runtime.h>
#include <hip/hip_bf16.h>
#include <math.h>

#define N_TOPICS 128
#define N_WORDS  32000
#define BATCH    64
#define EPS      1e-10f

typedef __attribute__((ext_vector_type(2))) float v2f;
typedef __attribute__((ext_vector_type(8))) float v8f;

// ---------------------------------------------------------------------------
// Kernel 1: per-topic softmax stats.  For each topic k (one block per topic):
//   max_k = max_w beta[k,w];  rsum_k = 1 / sum_w exp(beta[k,w] - max_k)
// Stored in ws: ws[0..127] = max, ws[128..255] = rsum.
// ---------------------------------------------------------------------------
__device__ __forceinline__ float waveMax(float v) {
    #pragma unroll
    for (int o = 16; o > 0; o >>= 1) v = fmaxf(v, __shfl_xor(v, o, 32));
    return v;
}
__device__ __forceinline__ float waveSum(float v) {
    #pragma unroll
    for (int o = 16; o > 0; o >>= 1) v += __shfl_xor(v, o, 32);
    return v;
}

__global__ void topic_stats_kernel(const float* __restrict__ beta,
                                   float* __restrict__ ws) {
    const int k   = blockIdx.x;                 // 0..127
    const int tid = threadIdx.x;                // 0..255 (8 waves)
    const float* row = beta + (size_t)k * N_WORDS;

    __shared__ float red[8];
    __shared__ float bmax;

    // pass 1: max
    float m = -INFINITY;
    for (int i = tid; i < N_WORDS; i += 256) m = fmaxf(m, row[i]);
    m = waveMax(m);
    if ((tid & 31) == 0) red[tid >> 5] = m;
    __syncthreads();
    if (tid == 0) {
        float mm = red[0];
        #pragma unroll
        for (int w = 1; w < 8; ++w) mm = fmaxf(mm, red[w]);
        bmax = mm;
    }
    __syncthreads();
    const float mk = bmax;

    // pass 2: sum of exp(x - max)
    float s = 0.0f;
    for (int i = tid; i < N_WORDS; i += 256) s += __expf(row[i] - mk);
    s = waveSum(s);
    __syncthreads();
    if ((tid & 31) == 0) red[tid >> 5] = s;
    __syncthreads();
    if (tid == 0) {
        float ss = 0.0f;
        #pragma unroll
        for (int w = 0; w < 8; ++w) ss += red[w];
        ws[k]            = mk;
        ws[N_TOPICS + k] = 1.0f / ss;
    }
}

// ---------------------------------------------------------------------------
// Kernel 2: fused softmax-GEMM.
//   out[b,w] = sum_k ((theta[b,k]+EPS) * rsum_k) * exp(beta[k,w] - max_k)
// Block tile: M=64 (all of batch) x N=64, K=128 fully resident in LDS.
// 8 waves: wave = mTile(0..3) + 4*nHalf(0..1); each wave does one 16-row
// M tile x two 16-col N tiles with V_WMMA_F32_16X16X4_F32, K-loop step 4.
//
// Both LDS tiles are K-contiguous with stride 132 (== 4 mod 64), so every
// fragment is a single conflict-free ds_load_b64: lower half-wave reads
// K = k0,k0+1 (banks {4c,4c+1}), upper half-wave reads K = k0+2,k0+3
// (banks {4c+2,4c+3}) -> all 64 banks hit exactly once per instruction.
// ---------------------------------------------------------------------------
#define K_STRIDE 132   // 128 + 4 pad

__global__ void __launch_bounds__(256)
topic_gemm_kernel(const float* __restrict__ theta,
                  const float* __restrict__ beta,
                  const float* __restrict__ ws,
                  float* __restrict__ out) {
    __shared__ float A_lds[BATCH * K_STRIDE];   // [b][k]  33792 B
    __shared__ float B_lds[64 * K_STRIDE];      // [n][k]  33792 B (transposed!)

    const int tid   = threadIdx.x;
    const int nBase = blockIdx.x * 64;

    // Stage A = (theta + eps) * rsum_k   (64 x 128, theta is row-major [b][k])
    for (int i = tid; i < BATCH * N_TOPICS; i += 256) {
        const int b = i >> 7;
        const int k = i & 127;
        A_lds[b * K_STRIDE + k] = (theta[i] + EPS) * ws[N_TOPICS + k];
    }
    // Stage B^T = exp(beta - max_k): global reads coalesced along n,
    // LDS writes transposed to [n][k].
    for (int i = tid; i < N_TOPICS * 64; i += 256) {
        const int k = i >> 6;
        const int n = i & 63;
        B_lds[n * K_STRIDE + k] =
            __expf(beta[(size_t)k * N_WORDS + nBase + n] - ws[k]);
    }
    __syncthreads();

    const int wave  = tid >> 5;
    const int lane  = tid & 31;
    const int mTile = wave & 3;        // 0..3 -> rows mTile*16..+15
    const int nHalf = wave >> 2;       // 0..1 -> cols nHalf*32..+31

    const int row  = mTile * 16 + (lane & 15);
    const int kOff = (lane < 16) ? 0 : 2;       // ISA 16x4 f32 A/B layout
    const int col0 = nHalf * 32 + (lane & 15);
    const int col1 = col0 + 16;

    const float* aRow  = &A_lds[row * K_STRIDE + kOff];
    const float* bRow0 = &B_lds[col0 * K_STRIDE + kOff];
    const float* bRow1 = &B_lds[col1 * K_STRIDE + kOff];

    v8f acc0 = {};
    v8f acc1 = {};

    #pragma unroll
    for (int k0 = 0; k0 < N_TOPICS; k0 += 4) {
        // Each fragment: {K=k0+kOff, K=k0+kOff+1} contiguous -> one b64 load.
        v2f a  = *(const v2f*)(aRow  + k0);
        v2f b0 = *(const v2f*)(bRow0 + k0);
        v2f b1 = *(const v2f*)(bRow1 + k0);

        acc0 = __builtin_amdgcn_wmma_f32_16x16x4_f32(
            false, a, false, b0, (short)0, acc0, false, false);
        acc1 = __builtin_amdgcn_wmma_f32_16x16x4_f32(
            false, a, false, b1, (short)0, acc1, false, false);
    }

    // Store: C/D 16x16 f32 layout — VGPR v: lanes 0-15 -> M=v, lanes 16-31 -> M=v+8
    const int rbase = mTile * 16 + ((lane >> 4) << 3);
    const int ncol  = nBase + nHalf * 32 + (lane & 15);
    #pragma unroll
    for (int v = 0; v < 8; ++v) {
        const size_t r = (size_t)(rbase + v);
        out[r * N_WORDS + ncol]      = acc0[v];
        out[r * N_WORDS + ncol + 16] = acc1[v];
    }
}

// ---------------------------------------------------------------------------
extern "C" void kernel_launch(void* const* d_in, const int* in_sizes, int n_in,
                              void* d_out, int out_size, void* d_ws, size_t ws_size,
                              hipStream_t stream) {
    const float* theta = (const float*)d_in[0];   // (64, 128)
    const float* beta  = (const float*)d_in[1];   // (128, 32000)
    float*       out   = (float*)d_out;           // (64, 32000)
    float*       ws    = (float*)d_ws;            // 256 floats of stats

    topic_stats_kernel<<<N_TOPICS, 256, 0, stream>>>(beta, ws);
    topic_gemm_kernel<<<N_WORDS / 64, 256, 0, stream>>>(theta, beta, ws, out);
}